// UniversalP_43748536877624
// MI455X (gfx1250) — compile-verified
//
#include <hip/hip_runtime.h>
#include <cstddef>
#include <cstdint>

#define NN      100000
#define EE      3200000
#define FEATS   128
#define CLASSES 8
#define HIDDEN  64
#define HID2    139   // 3 + CLASSES + FEATS
#define PSTRIDE 144   // HID2 padded to 9*16 tiles
#define MT16    (NN / 16)   // 6250 M-tiles of 16 rows

typedef float v2f __attribute__((ext_vector_type(2)));
typedef float v8f __attribute__((ext_vector_type(8)));

// ---------------------------------------------------------------------------
// Tiled GEMM, K fixed at 128: Out[M x NT*16] = X[M x 128] @ B[128 x NT*16]
//  - 256 threads = 8 wave32; each wave owns one 16-row M-tile and keeps all
//    NT 16x16 accumulator tiles in VGPRs (A fragment loaded once, reused NT x).
//  - B staged one 32-row K-chunk at a time into LDS via the gfx1250 async-DMA
//    path (global_load_async_to_lds_b32 + s_wait_asynccnt), stored row-pair
//    interleaved so each WMMA B-fragment is ONE aligned ds_load_b64:
//      word( K, col ) = (K>>1)*PITCH + 2*col + (K&1)
//    PITCH = 2*LDBP with LDBP = 16 mod 32 -> lane halves (K vs K+2) land on
//    disjoint bank halves 0-31 / 32-63: conflict-free b64 reads.
//  - B must have exactly NT*16 columns, contiguous (pre-packed if needed).
// ---------------------------------------------------------------------------
template <int NT>
__global__ __launch_bounds__(256) void k_gemm128_wmma(
    const float* __restrict__ X, const float* __restrict__ B,
    const float* __restrict__ bias, float* __restrict__ Out,
    int ldOut, int doRelu)
{
    constexpr int LDB   = NT * 16;                          // B columns (packed)
    constexpr int LDBP  = (LDB % 32 == 16) ? LDB : LDB + 16; // bank-shift pad
    constexpr int PITCH = 2 * LDBP;                          // words / row-pair
    __shared__ __align__(16) float Bs[16 * PITCH];

    const int lane = threadIdx.x & 31;
    const int wave = threadIdx.x >> 5;                 // 0..7
    const int m0   = (blockIdx.x * 8 + wave) * 16;     // wave's M-tile origin
    const int nl   = lane & 15;
    const int h2   = (lane >> 4) * 2;                  // K sub-offset 0 / 2
    const size_t arow = (size_t)(m0 + nl) * FEATS;     // A row for this lane
    const uint64_t Bg = (uint64_t)B;

    v8f acc[NT];
#pragma unroll
    for (int i = 0; i < NT; ++i) acc[i] = {};

    for (int kc = 0; kc < 128; kc += 32) {
        __syncthreads();   // previous chunk fully consumed by all waves
        // ---- async-DMA stage: B[kc..kc+32)[0..LDB) -> LDS, interleaved ----
        for (int idx = threadIdx.x; idx < 32 * LDB; idx += 256) {
            const int kr = idx / LDB, j = idx % LDB;
            const uint32_t lds_off =
                (uint32_t)(size_t)&Bs[(kr >> 1) * PITCH + 2 * j + (kr & 1)];
            const uint32_t voff = (uint32_t)((kc * LDB + idx) * sizeof(float));
            asm volatile("global_load_async_to_lds_b32 %0, %1, %2"
                         :: "v"(lds_off), "v"(voff), "s"(Bg) : "memory");
        }
        asm volatile("s_wait_asynccnt 0" ::: "memory");
        __syncthreads();   // all waves' staging visible

        if (m0 < NN) {     // wave-uniform guard for the ragged last block
#pragma unroll
            for (int kk = 0; kk < 32; kk += 4) {
                // A 16x4 f32 frag: VGPR0=K(k0), VGPR1=K(k0+1)  (ISA 7.12.2)
                const v2f a = *(const v2f*)(X + arow + kc + kk + h2);
                const int kp = (kk + h2) >> 1;     // row-pair index (k0 even)
#pragma unroll
                for (int nt = 0; nt < NT; ++nt) {
                    const v2f b =
                        *(const v2f*)&Bs[kp * PITCH + (nt * 16 + nl) * 2];
                    acc[nt] = __builtin_amdgcn_wmma_f32_16x16x4_f32(
                        false, a, false, b, (short)0, acc[nt], false, false);
                }
            }
        }
    }

    if (m0 >= NN) return;
    const int rbase = m0 + ((lane >> 4) ? 8 : 0);      // C/D: M 0-7 | 8-15 halves
#pragma unroll
    for (int nt = 0; nt < NT; ++nt) {
        const int col = nt * 16 + nl;
        const float bv = bias ? bias[col] : 0.0f;
#pragma unroll
        for (int r = 0; r < 8; ++r) {
            float v = acc[nt][r] + bv;
            if (doRelu) v = fmaxf(v, 0.0f);
            Out[(size_t)(rbase + r) * ldOut + col] = v;
        }
    }
}

// ---------------------------------------------------------------------------
// Utility / graph kernels
// ---------------------------------------------------------------------------
__global__ void k_fill(float* p, float v, int n) {
    int t = blockIdx.x * blockDim.x + threadIdx.x;
    if (t < n) p[t] = v;
}

__global__ void k_degcount(const int* __restrict__ dst, float* deg, int e) {
    int t = blockIdx.x * blockDim.x + threadIdx.x;
    if (t < e) unsafeAtomicAdd(&deg[dst[t]], 1.0f);
}

__global__ void k_invsqrt(const float* __restrict__ deg, float* isq, int n) {
    int t = blockIdx.x * blockDim.x + threadIdx.x;
    if (t < n) { float d = deg[t]; isq[t] = (d > 0.0f) ? rsqrtf(d) : 0.0f; }
}

__global__ void k_enorm(const int* __restrict__ src, const int* __restrict__ dst,
                        const float* __restrict__ isq, float* en, int e) {
    int t = blockIdx.x * blockDim.x + threadIdx.x;
    if (t < e) en[t] = isq[src[t]] * isq[dst[t]];
}

// zero-padded pack: Bpad[128 x 144] = Wa1[9:137, 0:139] | 0
__global__ void k_packB(const float* __restrict__ Wa1, float* __restrict__ Bp) {
    int t = blockIdx.x * blockDim.x + threadIdx.x;
    if (t >= 128 * PSTRIDE) return;
    int k = t / PSTRIDE, j = t % PSTRIDE;
    Bp[t] = (j < HID2) ? Wa1[(size_t)(9 + k) * HID2 + j] : 0.0f;
}

// h = h1 @ W2 + b2   (64 -> 8, tiny: one thread per (node,class))
__global__ void k_gemm2(const float* __restrict__ h1, const float* __restrict__ W2,
                        const float* __restrict__ b2, float* __restrict__ h, int n8) {
    int t = blockIdx.x * blockDim.x + threadIdx.x;
    if (t >= n8) return;
    int n = t >> 3, c = t & 7;
    float a = b2[c];
    const float* hr = h1 + (size_t)n * HIDDEN;
#pragma unroll 8
    for (int k = 0; k < HIDDEN; ++k) a += hr[k] * W2[k * CLASSES + c];
    h[t] = a;
}

// nxt[n,c] = isq[n]^2 * cur[n,c]  (self-loop seed), float4-vectorized
__global__ void k_selfinit4(float4* __restrict__ nxt, const float4* __restrict__ cur,
                            const float* __restrict__ isq, int n4) {
    int t = blockIdx.x * blockDim.x + threadIdx.x;
    if (t >= n4) return;
    float s = isq[t >> 1];          // 2 float4 per node row of 8
    float s2 = s * s;
    float4 v = cur[t];
    nxt[t] = make_float4(s2 * v.x, s2 * v.y, s2 * v.z, s2 * v.w);
}

// scatter: nxt[dst,:] += norm[e] * cur[src,:] ; one thread per edge
__global__ void k_edge(const int* __restrict__ src, const int* __restrict__ dst,
                       const float* __restrict__ en, const float* __restrict__ cur,
                       float* __restrict__ nxt, int e) {
    int t = blockIdx.x * blockDim.x + threadIdx.x;
    if (t >= e) return;
    const float w = en[t];
    const float4* cr = (const float4*)(cur + (size_t)src[t] * CLASSES);
    float4 v0 = cr[0], v1 = cr[1];
    float* op = nxt + (size_t)dst[t] * CLASSES;
    unsafeAtomicAdd(op + 0, w * v0.x);
    unsafeAtomicAdd(op + 1, w * v0.y);
    unsafeAtomicAdd(op + 2, w * v0.z);
    unsafeAtomicAdd(op + 3, w * v0.w);
    unsafeAtomicAdd(op + 4, w * v1.x);
    unsafeAtomicAdd(op + 5, w * v1.y);
    unsafeAtomicAdd(op + 6, w * v1.z);
    unsafeAtomicAdd(op + 7, w * v1.w);
}

__global__ void k_axpy4(float4* __restrict__ acc, const float4* __restrict__ v,
                        float d, int n4) {
    int t = blockIdx.x * blockDim.x + threadIdx.x;
    if (t >= n4) return;
    float4 a = acc[t], b = v[t];
    acc[t] = make_float4(a.x + d * b.x, a.y + d * b.y, a.z + d * b.z, a.w + d * b.w);
}

__global__ void k_scale4(float4* __restrict__ dst, const float4* __restrict__ s,
                         float f, int n4) {
    int t = blockIdx.x * blockDim.x + threadIdx.x;
    if (t >= n4) return;
    float4 v = s[t];
    dst[t] = make_float4(f * v.x, f * v.y, f * v.z, f * v.w);
}

// bclass[c,j] = Wa1[1+c, j] + ba1[j]
__global__ void k_bclass(const float* __restrict__ Wa1, const float* __restrict__ ba1,
                         float* __restrict__ bc) {
    int t = blockIdx.x * blockDim.x + threadIdx.x;
    if (t >= CLASSES * HID2) return;
    int c = t / HID2, j = t % HID2;
    bc[t] = Wa1[(size_t)(1 + c) * HID2 + j] + ba1[j];
}

// out[n,c] = relu(P[n,:] + h[n,c]*Wa1[0,:] + bclass[c,:]) . Wa2 + ba2
__global__ void k_stage2(const float* __restrict__ hd1, const float* __restrict__ P,
                         const float* __restrict__ Wa1, const float* __restrict__ bc,
                         const float* __restrict__ Wa2, const float* __restrict__ ba2,
                         float* __restrict__ out, int n8) {
    int t = blockIdx.x * blockDim.x + threadIdx.x;
    if (t >= n8) return;
    int n = t >> 3, c = t & 7;
    float s = hd1[t];
    const float* prow = P + (size_t)n * PSTRIDE;
    const float* bcr  = bc + (size_t)c * HID2;
    float acc = 0.0f;
    for (int j = 0; j < HID2; ++j) {
        float hid = fmaxf(prow[j] + s * Wa1[j] + bcr[j], 0.0f);
        acc += hid * Wa2[j];
    }
    out[t] = acc + ba2[0];
}

// ---------------------------------------------------------------------------
extern "C" void kernel_launch(void* const* d_in, const int* in_sizes, int n_in,
                              void* d_out, int out_size, void* d_ws, size_t ws_size,
                              hipStream_t stream) {
    const float* x   = (const float*)d_in[0];
    const int*   ei  = (const int*)  d_in[1];
    const float* W1  = (const float*)d_in[2];
    const float* b1  = (const float*)d_in[3];
    const float* W2  = (const float*)d_in[4];
    const float* b2  = (const float*)d_in[5];
    const float* Wa1 = (const float*)d_in[6];
    const float* ba1 = (const float*)d_in[7];
    const float* Wa2 = (const float*)d_in[8];
    const float* ba2 = (const float*)d_in[9];
    const int* srcI = ei;
    const int* dstI = ei + EE;

    // carve workspace (float cursor, 256B aligned chunks)
    float* ws = (float*)d_ws;
    size_t cur = 0;
    auto alloc = [&](size_t nfl) {
        float* p = ws + cur;
        cur += (nfl + 63) & ~(size_t)63;
        return p;
    };
    float* deg    = alloc(NN);
    float* isq    = alloc(NN);
    float* enorm  = alloc(EE);
    float* h1     = alloc((size_t)NN * HIDDEN);
    float* hA     = alloc((size_t)NN * CLASSES);
    float* hB     = alloc((size_t)NN * CLASSES);
    float* hacc   = alloc((size_t)NN * CLASSES);
    float* hd1    = alloc((size_t)NN * CLASSES);
    float* P      = alloc((size_t)NN * PSTRIDE);
    float* Bpad   = alloc((size_t)128 * PSTRIDE);
    float* bclass = alloc((size_t)CLASSES * HID2);

    const int B = 256;
    auto gb = [](long n, int b) { return (unsigned)((n + b - 1) / b); };
    const int N8 = NN * CLASSES;
    const int N4 = N8 / 4;                        // float4 elements
    const unsigned gemmBlocks = (MT16 + 7) / 8;   // 8 M-tiles per block

    double ds = 1.0, p9 = 1.0;
    for (int l = 0; l < 10; ++l) { p9 *= 0.9; ds += p9; }
    const float invDS = (float)(1.0 / ds);

    // graph norm: deg = 1 + indeg ; isq = rsqrt ; per-edge norm
    k_fill    <<<gb(NN, B), B, 0, stream>>>(deg, 1.0f, NN);
    k_degcount<<<gb(EE, B), B, 0, stream>>>(dstI, deg, EE);
    k_invsqrt <<<gb(NN, B), B, 0, stream>>>(deg, isq, NN);
    k_enorm   <<<gb(EE, B), B, 0, stream>>>(srcI, dstI, isq, enorm, EE);

    // stage 1: h1 = relu(x@W1+b1)  [WMMA, NT=4] ; hA = h1@W2+b2
    k_gemm128_wmma<4><<<gemmBlocks, 256, 0, stream>>>(x, W1, b1, h1, HIDDEN, 1);
    k_gemm2<<<gb(N8, B), B, 0, stream>>>(h1, W2, b2, hA, N8);

    // APPNP diffusion: hacc = sum_l 0.9^l A^l h  (ping-pong curp/nxtp)
    auto diffuse = [&](float* curp, float* nxtp) {
        k_scale4<<<gb(N4, B), B, 0, stream>>>((float4*)hacc, (const float4*)curp,
                                              1.0f, N4);
        float d = 1.0f;
        for (int l = 0; l < 10; ++l) {
            d *= 0.9f;
            k_selfinit4<<<gb(N4, B), B, 0, stream>>>((float4*)nxtp,
                                                     (const float4*)curp, isq, N4);
            k_edge<<<gb(EE, B), B, 0, stream>>>(srcI, dstI, enorm, curp, nxtp, EE);
            k_axpy4<<<gb(N4, B), B, 0, stream>>>((float4*)hacc,
                                                 (const float4*)nxtp, d, N4);
            float* tmp = curp; curp = nxtp; nxtp = tmp;
        }
    };

    diffuse(hA, hB);
    k_scale4<<<gb(N4, B), B, 0, stream>>>((float4*)hd1, (const float4*)hacc,
                                          invDS, N4);

    // stage 2: P = x @ Wa1[9:,:] (zero-padded to 144 cols)  [WMMA, NT=9]
    k_bclass<<<gb(CLASSES * HID2, B), B, 0, stream>>>(Wa1, ba1, bclass);
    k_packB <<<gb(128 * PSTRIDE, B), B, 0, stream>>>(Wa1, Bpad);
    k_gemm128_wmma<9><<<gemmBlocks, 256, 0, stream>>>(x, Bpad, nullptr, P, PSTRIDE, 0);
    k_stage2<<<gb(N8, B), B, 0, stream>>>(hd1, P, Wa1, bclass, Wa2, ba2, hA, N8);

    // second diffusion + final scale into d_out
    diffuse(hA, hB);
    k_scale4<<<gb(N4, B), B, 0, stream>>>((float4*)d_out, (const float4*)hacc,
                                          invDS, N4);
}